// UNOBlock_41437844471948
// MI455X (gfx1250) — compile-verified
//
#include <hip/hip_runtime.h>
#include <hip/hip_bf16.h>

#define B_SZ   8
#define C_CH   192
#define EMB_D  768
#define HW_N   1024
#define GROUPS 32
#define CPG    6            // C_CH / GROUPS
#define NMODE  512          // 32 rows * 16 cols kept
#define TWO_PI 6.2831853071795864f
#define SKIP_SCALE_C 1.0f

typedef __attribute__((ext_vector_type(2))) float v2f;
typedef __attribute__((ext_vector_type(8))) float v8f;

__device__ __forceinline__ v8f wmma4(v2f a, v2f b, v8f c) {
  // V_WMMA_F32_16X16X4_F32 : D(16x16,f32) = A(16x4,f32) x B(4x16,f32) + C
  return __builtin_amdgcn_wmma_f32_16x16x4_f32(false, a, false, b, (short)0, c, false, false);
}

__device__ __forceinline__ int imin(int a, int b) { return a < b ? a : b; }

// ---------------------------------------------------------------------------
// Branch-free inner product loop: K%4==0 assumed, pointers pre-offset per lane.
// A2: A k-stride==1 (load b64); B2: B k-stride==1 (load b64).
// ---------------------------------------------------------------------------
template<bool A2, bool B2>
__device__ __forceinline__ void gemm_inner(const float* __restrict__ pA, long long csA,
                                           const float* __restrict__ pB0,
                                           const float* __restrict__ pB1, long long rsB,
                                           int K, v8f& acc0, v8f& acc1)
{
  const long long stepA = 4 * csA;
  const long long stepB = 4 * rsB;
#pragma unroll 4
  for (int k0 = 0; k0 < K; k0 += 4) {
    v2f av, b0, b1;
    if (A2) { av = *(const v2f*)pA; }
    else    { av.x = pA[0]; av.y = pA[csA]; }
    if (B2) { b0 = *(const v2f*)pB0; b1 = *(const v2f*)pB1; }
    else    { b0.x = pB0[0]; b0.y = pB0[rsB]; b1.x = pB1[0]; b1.y = pB1[rsB]; }
    acc0 = wmma4(av, b0, acc0);
    acc1 = wmma4(av, b1, acc1);
    pA += stepA; pB0 += stepB; pB1 += stepB;
  }
}

// ---------------------------------------------------------------------------
// Generic strided batched GEMM: C[z] = post*(alpha*A[z]@B[z] + bias + resid)
// One wave per 16x32 output tile (2 accumulators share each A fragment).
// OOB handled by clamping load indices (stores are guarded), so the hot loop
// is branch-free and EXEC stays all-ones for every WMMA.
// ---------------------------------------------------------------------------
__global__ void k_gemm_wmma(
    const float* __restrict__ A, long long baseA, long long zsA, long long rsA, long long csA,
    const float* __restrict__ Bm, long long baseB, long long zsB, long long rsB, long long csB,
    float* __restrict__ Cm, long long baseC, long long zsC, long long rsC,
    const float* __restrict__ bias, int bias_mode,            // 0 none, 1 per-m, 2 per-n
    const float* __restrict__ resid, long long baseR, long long zsR, long long rsR,
    float alpha, float post_scale, int M, int N, int K)
{
  const int lane = (int)threadIdx.x;          // wave32
  const int half = lane >> 4;                 // 0 | 1
  const int l16  = lane & 15;
  const int tn = (int)blockIdx.x * 32;
  const int tm = (int)blockIdx.y * 16;
  const long long z = (long long)blockIdx.z;

  const float* Ab = A  + baseA + z * zsA;
  const float* Bb = Bm + baseB + z * zsB;

  const int am  = imin(tm + l16, M - 1);          // clamped load row
  const int bn0 = imin(tn + l16, N - 1);          // clamped load cols
  const int bn1 = imin(tn + 16 + l16, N - 1);

  const float* pA  = Ab + (long long)am * rsA + (long long)(half * 2) * csA;
  const float* pB0 = Bb + (long long)(half * 2) * rsB + (long long)bn0 * csB;
  const float* pB1 = Bb + (long long)(half * 2) * rsB + (long long)bn1 * csB;

  v8f acc0 = {0.f,0.f,0.f,0.f,0.f,0.f,0.f,0.f};
  v8f acc1 = {0.f,0.f,0.f,0.f,0.f,0.f,0.f,0.f};

  const bool a2 = (csA == 1);
  const bool b2 = (rsB == 1);
  if      ( a2 &&  b2) gemm_inner<true , true >(pA, csA, pB0, pB1, rsB, K, acc0, acc1);
  else if ( a2 && !b2) gemm_inner<true , false>(pA, csA, pB0, pB1, rsB, K, acc0, acc1);
  else if (!a2 &&  b2) gemm_inner<false, true >(pA, csA, pB0, pB1, rsB, K, acc0, acc1);
  else                 gemm_inner<false, false>(pA, csA, pB0, pB1, rsB, K, acc0, acc1);

  float* Cb = Cm + baseC + z * zsC;
  const float* Rb = resid ? (resid + baseR + z * zsR) : nullptr;
#pragma unroll
  for (int t = 0; t < 2; ++t) {
    const int n = tn + t * 16 + l16;
    if (n >= N) continue;
    const v8f& acc = t ? acc1 : acc0;
    const float bnb = (bias_mode == 2) ? bias[n] : 0.f;
#pragma unroll
    for (int v = 0; v < 8; ++v) {
      const int m = tm + v + half * 8;          // C/D layout: row = v (+8 hi half)
      if (m < M) {
        float val = acc[v] * alpha;
        val += (bias_mode == 1) ? bias[m] : bnb;
        if (Rb) val += Rb[(long long)m * rsR + n];
        Cb[(long long)m * rsC + n] = val * post_scale;
      }
    }
  }
}

// ---------------------------------------------------------------------------
// GroupNorm (mode 0: +SiLU, mode 1: +FiLM(par)+SiLU, mode 2: plain)
// ---------------------------------------------------------------------------
__global__ void k_groupnorm(const float* __restrict__ src, float* __restrict__ dst,
                            const float* __restrict__ w, const float* __restrict__ bgn,
                            const float* __restrict__ par, int mode)
{
  const int bg = (int)blockIdx.x;
  const int b = bg / GROUPS, g = bg % GROUPS;
  const long long base = ((long long)b * C_CH + g * CPG) * HW_N;
  const int Nel = CPG * HW_N;
  __shared__ float s1[256], s2[256];
  float sum = 0.f, ssq = 0.f;
  for (int i = threadIdx.x; i < Nel; i += blockDim.x) {
    float x = src[base + i]; sum += x; ssq += x * x;
  }
  s1[threadIdx.x] = sum; s2[threadIdx.x] = ssq; __syncthreads();
  for (int st = 128; st > 0; st >>= 1) {
    if ((int)threadIdx.x < st) { s1[threadIdx.x] += s1[threadIdx.x + st]; s2[threadIdx.x] += s2[threadIdx.x + st]; }
    __syncthreads();
  }
  const float mu  = s1[0] / (float)Nel;
  const float var = s2[0] / (float)Nel - mu * mu;
  const float rs  = rsqrtf(var + 1e-5f);
  for (int i = threadIdx.x; i < Nel; i += blockDim.x) {
    const int c = g * CPG + i / HW_N;
    float y = (src[base + i] - mu) * rs * w[c] + bgn[c];
    if (mode == 1) {
      const float scale = par[b * (2 * C_CH) + c] + 1.f;
      const float shift = par[b * (2 * C_CH) + C_CH + c];
      y = shift + y * scale;
    }
    if (mode != 2) y = y / (1.f + __expf(-y));   // SiLU
    dst[base + i] = y;
  }
}

// ---------------------------------------------------------------------------
// Forward 2D DFT, keeping cols 0..15 of rfft: XF[p,b,c,u,v], p=re/im
// ---------------------------------------------------------------------------
__global__ void k_dft_fwd(const float* __restrict__ src, float* __restrict__ XF)
{
  const int bc = (int)blockIdx.x;
  __shared__ float xs[1024];
  __shared__ float tr[512], ti[512];
  const float* sp = src + (long long)bc * HW_N;
  for (int i = threadIdx.x; i < 1024; i += blockDim.x) xs[i] = sp[i];
  __syncthreads();
  for (int idx = threadIdx.x; idx < 512; idx += blockDim.x) {
    const int h = idx >> 4, v = idx & 15;
    float sr = 0.f, si = 0.f;
    for (int w = 0; w < 32; ++w) {
      float sn, cs; __sincosf(-TWO_PI * (float)((v * w) & 31) * (1.f / 32.f), &sn, &cs);
      const float xv = xs[h * 32 + w];
      sr += xv * cs; si += xv * sn;
    }
    tr[idx] = sr; ti[idx] = si;
  }
  __syncthreads();
  for (int idx = threadIdx.x; idx < 512; idx += blockDim.x) {
    const int u = idx >> 4, v = idx & 15;
    float sr = 0.f, si = 0.f;
    for (int h = 0; h < 32; ++h) {
      float sn, cs; __sincosf(-TWO_PI * (float)((u * h) & 31) * (1.f / 32.f), &sn, &cs);
      const float a = tr[h * 16 + v], bb = ti[h * 16 + v];
      sr += a * cs - bb * sn;
      si += a * sn + bb * cs;
    }
    const long long base = (long long)bc * NMODE + idx;        // p=0 plane
    XF[base] = sr;
    XF[base + (long long)B_SZ * C_CH * NMODE] = si;            // p=1 plane
  }
}

// ---------------------------------------------------------------------------
// Per-mode complex channel mix via WMMA, dual 16-col tiles per wave.
// A_stack (16 x 192): rows 0..7 = Re(XF[b,:]), 8..15 = Im(XF[b,:]) per mode.
// AR = A@Wr, AI = A@Wi, recombine Re/Im with a lane^16 shuffle.
// grid: (512 modes, 6 n-tiles), block 32 (one wave).
// ---------------------------------------------------------------------------
__global__ void k_spectral_mix(const float* __restrict__ XF,
                               const float* __restrict__ w1r, const float* __restrict__ w1i,
                               const float* __restrict__ w2r, const float* __restrict__ w2i,
                               float* __restrict__ OF)
{
  const int mode = (int)blockIdx.x;           // u*16 + v
  const int tn   = (int)blockIdx.y * 32;
  const int lane = (int)threadIdx.x;
  const int half = lane >> 4, l16 = lane & 15;
  const int u = mode >> 4, v = mode & 15;
  const float* wr = (u < 16) ? w1r : w2r;     // rows 0..15 -> w1, 16..31 -> w2
  const float* wi = (u < 16) ? w1i : w2i;
  const int woff = (u & 15) * 16 + v;         // w[i,o,u',v] tail index
  const int n0 = tn + l16;
  const int n1 = n0 + 16;

  const long long wk = (long long)C_CH * 256; // k (input-channel) stride in w
  const float* pX  = XF + ((long long)l16 * C_CH + half * 2) * NMODE + mode;
  const float* pR0 = wr + ((long long)(half * 2) * C_CH + n0) * 256 + woff;
  const float* pR1 = wr + ((long long)(half * 2) * C_CH + n1) * 256 + woff;
  const float* pI0 = wi + ((long long)(half * 2) * C_CH + n0) * 256 + woff;
  const float* pI1 = wi + ((long long)(half * 2) * C_CH + n1) * 256 + woff;

  v8f accR0 = {0.f,0.f,0.f,0.f,0.f,0.f,0.f,0.f};
  v8f accR1 = accR0, accI0 = accR0, accI1 = accR0;

#pragma unroll 4
  for (int k0 = 0; k0 < C_CH; k0 += 4) {
    v2f a, br0, br1, bi0, bi1;
    a.x   = pX[0];      a.y   = pX[NMODE];
    br0.x = pR0[0];     br0.y = pR0[wk];
    br1.x = pR1[0];     br1.y = pR1[wk];
    bi0.x = pI0[0];     bi0.y = pI0[wk];
    bi1.x = pI1[0];     bi1.y = pI1[wk];
    accR0 = wmma4(a, br0, accR0);
    accR1 = wmma4(a, br1, accR1);
    accI0 = wmma4(a, bi0, accI0);
    accI1 = wmma4(a, bi1, accI1);
    pX += 4 * NMODE; pR0 += 4 * wk; pR1 += 4 * wk; pI0 += 4 * wk; pI1 += 4 * wk;
  }

  // Re(out)[b] = AR[b] - AI[b+8] (lanes 0..15) ; Im(out)[b] = AI[b] + AR[b+8] (lanes 16..31)
  const int xl = lane ^ 16;
  const float sgn = (lane < 16) ? -1.f : 1.f;
#pragma unroll
  for (int vv = 0; vv < 8; ++vv) {
    const long long mrow = (long long)(half * 8 + vv);          // p*8+b
    const float r0 = accR0[vv] + sgn * __shfl(accI0[vv], xl, 32);
    const float r1 = accR1[vv] + sgn * __shfl(accI1[vv], xl, 32);
    OF[(mrow * C_CH + n0) * NMODE + mode] = r0;
    OF[(mrow * C_CH + n1) * NMODE + mode] = r1;
  }
}

// ---------------------------------------------------------------------------
// Inverse: ifft along u (32-pt complex), irfft along w (bins 0..15, bin16=0).
// ---------------------------------------------------------------------------
__global__ void k_dft_inv(const float* __restrict__ OF, const float* __restrict__ bias,
                          const float* __restrict__ skip, float skip_scale,
                          float* __restrict__ dst)
{
  const int bo = (int)blockIdx.x;
  const int o = bo % C_CH;
  __shared__ float yr[512], yi[512];
  const long long base0 = (long long)bo * NMODE;
  const long long imoff = (long long)B_SZ * C_CH * NMODE;
  for (int idx = threadIdx.x; idx < 512; idx += blockDim.x) {
    const int h = idx >> 4, v = idx & 15;
    float sr = 0.f, si = 0.f;
    for (int u = 0; u < 32; ++u) {
      float sn, cs; __sincosf(TWO_PI * (float)((u * h) & 31) * (1.f / 32.f), &sn, &cs);
      const float fr = OF[base0 + u * 16 + v];
      const float fi = OF[base0 + imoff + u * 16 + v];
      sr += fr * cs - fi * sn;
      si += fr * sn + fi * cs;
    }
    yr[idx] = sr * (1.f / 32.f);
    yi[idx] = si * (1.f / 32.f);
  }
  __syncthreads();
  const float ob = bias[o];
  float* dp = dst + (long long)bo * HW_N;
  const float* sp = skip ? (skip + (long long)bo * HW_N) : nullptr;
  for (int idx = threadIdx.x; idx < 1024; idx += blockDim.x) {
    const int h = idx >> 5, w = idx & 31;
    float acc = yr[h * 16];                   // bin 0: real part only (irfft)
    for (int v = 1; v < 16; ++v) {
      float sn, cs; __sincosf(TWO_PI * (float)((v * w) & 31) * (1.f / 32.f), &sn, &cs);
      acc += 2.f * (yr[h * 16 + v] * cs - yi[h * 16 + v] * sn);
    }
    acc = acc * (1.f / 32.f) + ob;
    if (sp) acc = (acc + sp[idx]) * skip_scale;
    dp[idx] = acc;
  }
}

// ---------------------------------------------------------------------------
// Row softmax over 1024 (one block per row)
// ---------------------------------------------------------------------------
__global__ void k_softmax(float* __restrict__ logits)
{
  float* p = logits + (long long)blockIdx.x * 1024;
  __shared__ float red[256];
  float m = -1e30f;
  for (int i = threadIdx.x; i < 1024; i += 256) m = fmaxf(m, p[i]);
  red[threadIdx.x] = m; __syncthreads();
  for (int st = 128; st > 0; st >>= 1) {
    if ((int)threadIdx.x < st) red[threadIdx.x] = fmaxf(red[threadIdx.x], red[threadIdx.x + st]);
    __syncthreads();
  }
  m = red[0]; __syncthreads();
  float s = 0.f;
  for (int i = threadIdx.x; i < 1024; i += 256) { float e = __expf(p[i] - m); p[i] = e; s += e; }
  red[threadIdx.x] = s; __syncthreads();
  for (int st = 128; st > 0; st >>= 1) {
    if ((int)threadIdx.x < st) red[threadIdx.x] += red[threadIdx.x + st];
    __syncthreads();
  }
  const float inv = 1.f / red[0];
  for (int i = threadIdx.x; i < 1024; i += 256) p[i] *= inv;
}

// ---------------------------------------------------------------------------
extern "C" void kernel_launch(void* const* d_in, const int* in_sizes, int n_in,
                              void* d_out, int out_size, void* d_ws, size_t ws_size,
                              hipStream_t stream)
{
  (void)in_sizes; (void)n_in; (void)out_size; (void)ws_size;
  const float* x      = (const float*)d_in[0];
  const float* emb    = (const float*)d_in[1];
  const float* gn0_w  = (const float*)d_in[2];
  const float* gn0_b  = (const float*)d_in[3];
  const float* c0_w1r = (const float*)d_in[4];
  const float* c0_w1i = (const float*)d_in[5];
  const float* c0_w2r = (const float*)d_in[6];
  const float* c0_w2i = (const float*)d_in[7];
  const float* c0_b   = (const float*)d_in[8];
  const float* aff_w  = (const float*)d_in[9];
  const float* aff_b  = (const float*)d_in[10];
  const float* gn1_w  = (const float*)d_in[11];
  const float* gn1_b  = (const float*)d_in[12];
  const float* c1_w1r = (const float*)d_in[13];
  const float* c1_w1i = (const float*)d_in[14];
  const float* c1_w2r = (const float*)d_in[15];
  const float* c1_w2i = (const float*)d_in[16];
  const float* c1_b   = (const float*)d_in[17];
  const float* gn2_w  = (const float*)d_in[18];
  const float* gn2_b  = (const float*)d_in[19];
  const float* qkv_w  = (const float*)d_in[20];
  const float* qkv_b  = (const float*)d_in[21];
  const float* proj_w = (const float*)d_in[22];
  const float* proj_b = (const float*)d_in[23];
  float* out = (float*)d_out;

  float* ws = (float*)d_ws;
  const long long NT = (long long)B_SZ * C_CH * HW_N;     // 1,572,864
  float* A_   = ws;                    // activations (a0 / a1 / hn)
  float* XF   = A_  + NT;              // forward spectrum [p,b,c,512]
  float* OF   = XF  + NT;              // mixed spectrum   [p,b,o,512]
  float* H0   = OF  + NT;              // spectral conv0 out
  float* Hb   = H0  + NT;              // h after skip (residual source)
  float* QKV  = Hb  + NT;              // [b, 576, 1024]
  float* LOG  = QKV + (long long)B_SZ * 576 * HW_N;       // 1024x1024
  float* ATT  = LOG + (long long)HW_N * HW_N;             // [b, 192, 1024]
  float* PAR  = ATT + NT;              // [8, 384]

  const dim3 wv(32, 1, 1);

  // ---- branch 1: GN0 + SiLU -> spectral conv 0 ----
  k_groupnorm<<<B_SZ * GROUPS, 256, 0, stream>>>(x, A_, gn0_w, gn0_b, nullptr, 0);
  k_dft_fwd  <<<B_SZ * C_CH, 256, 0, stream>>>(A_, XF);
  k_spectral_mix<<<dim3(NMODE, 6, 1), wv, 0, stream>>>(XF, c0_w1r, c0_w1i, c0_w2r, c0_w2i, OF);
  k_dft_inv  <<<B_SZ * C_CH, 256, 0, stream>>>(OF, c0_b, nullptr, 1.f, H0);

  // ---- FiLM params: PAR = emb @ aff_w.T + aff_b  (M=8, N=384, K=768) ----
  k_gemm_wmma<<<dim3(12, 1, 1), wv, 0, stream>>>(
      emb, 0, 0, EMB_D, 1,
      aff_w, 0, 0, 1, EMB_D,
      PAR, 0, 0, 2 * C_CH,
      aff_b, 2, nullptr, 0, 0, 0,
      1.f, 1.f, B_SZ, 2 * C_CH, EMB_D);

  // ---- GN1 + FiLM + SiLU -> spectral conv 1 (+ residual x, *SKIP) ----
  k_groupnorm<<<B_SZ * GROUPS, 256, 0, stream>>>(H0, A_, gn1_w, gn1_b, PAR, 1);
  k_dft_fwd  <<<B_SZ * C_CH, 256, 0, stream>>>(A_, XF);
  k_spectral_mix<<<dim3(NMODE, 6, 1), wv, 0, stream>>>(XF, c1_w1r, c1_w1i, c1_w2r, c1_w2i, OF);
  k_dft_inv  <<<B_SZ * C_CH, 256, 0, stream>>>(OF, c1_b, x, SKIP_SCALE_C, Hb);

  // ---- attention: GN2 -> qkv -> per-head softmax(QK^T/8)V -> proj ----
  k_groupnorm<<<B_SZ * GROUPS, 256, 0, stream>>>(Hb, A_, gn2_w, gn2_b, nullptr, 2);

  // QKV[b] = qkv_w(576x192) @ hn[b](192x1024) + qkv_b[m]
  k_gemm_wmma<<<dim3(32, 36, B_SZ), wv, 0, stream>>>(
      qkv_w, 0, 0, C_CH, 1,
      A_, 0, (long long)C_CH * HW_N, HW_N, 1,
      QKV, 0, 576LL * HW_N, HW_N,
      qkv_b, 1, nullptr, 0, 0, 0,
      1.f, 1.f, 576, HW_N, C_CH);

  // per head-batch n: q/k/v channel c lives at QKV[n*196608 + (3c+s)*1024 + t]
  for (int n = 0; n < B_SZ * 3; ++n) {
    const long long nb = (long long)n * 576 * HW_N / 3;   // n * 196608
    // logits = (Q^T K)/8 : M=N=1024, K=64
    k_gemm_wmma<<<dim3(32, 64, 1), wv, 0, stream>>>(
        QKV, nb,        0, 1, 3 * HW_N,
        QKV, nb + HW_N, 0, 3 * HW_N, 1,
        LOG, 0, 0, HW_N,
        nullptr, 0, nullptr, 0, 0, 0,
        0.125f, 1.f, HW_N, HW_N, 64);
    k_softmax<<<HW_N, 256, 0, stream>>>(LOG);
    // O[c,q] = V(64x1024) @ P^T(1024x1024) : M=64, N=1024, K=1024 (rsB==1 -> b64 B loads)
    k_gemm_wmma<<<dim3(32, 4, 1), wv, 0, stream>>>(
        QKV, nb + 2 * HW_N, 0, 3 * HW_N, 1,
        LOG, 0, 0, 1, HW_N,
        ATT, (long long)n * 64 * HW_N, 0, HW_N,
        nullptr, 0, nullptr, 0, 0, 0,
        1.f, 1.f, 64, HW_N, HW_N);
  }

  // out[b] = (proj_w(192x192) @ ATT[b] + proj_b[m] + Hb[b]) * SKIP
  k_gemm_wmma<<<dim3(32, 12, B_SZ), wv, 0, stream>>>(
      proj_w, 0, 0, C_CH, 1,
      ATT, 0, (long long)C_CH * HW_N, HW_N, 1,
      out, 0, (long long)C_CH * HW_N, HW_N,
      proj_b, 1, Hb, 0, (long long)C_CH * HW_N, HW_N,
      1.f, SKIP_SCALE_C, C_CH, HW_N, C_CH);
}